// L2Accuracy_15427522527883
// MI455X (gfx1250) — compile-verified
//
#include <hip/hip_runtime.h>
#include <hip/hip_bf16.h>

#define THREADS 256
#define TILE    1024
#define VPT     (TILE / THREADS)   // 4 vertices per thread
#define MAXG    16
#define MAXT    16

#if defined(__gfx1250__) && __has_builtin(__builtin_amdgcn_global_load_async_to_lds_b128)
#define HAVE_ASYNC_LDS 1
#else
#define HAVE_ASYNC_LDS 0
#endif

#if HAVE_ASYNC_LDS
typedef int v4i __attribute__((vector_size(16)));   // matches builtin's 4 x i32 pointee
__device__ __forceinline__ void async_copy_b128(const void* gsrc, void* ldst) {
    // b128 async DMA: (global src AS1, LDS dst AS3, imm offset, imm cpol)
    __builtin_amdgcn_global_load_async_to_lds_b128(
        (__attribute__((address_space(1))) v4i*)gsrc,
        (__attribute__((address_space(3))) v4i*)ldst, 0, 0);
}
#endif

// ---------------- kernel 0: zero the per-garment accumulators -------------
__global__ void l2acc_init_kernel(float* __restrict__ seg_sum, int bg) {
    int i = blockIdx.x * blockDim.x + threadIdx.x;
    if (i < bg) seg_sum[i] = 0.0f;
}

// ---------------- kernel 1: per-vertex L2 error + ragged segment sums -----
__global__ void l2acc_seg_sum_kernel(const float* __restrict__ pred,
                                     const float* __restrict__ target,
                                     const int*   __restrict__ idx,   // [B, G+1]
                                     float*       __restrict__ seg_sum, // [B*G]
                                     int N, int G, int tilesPerBatch) {
    __shared__ float sp[TILE * 3];       // 12 KB pred tile
    __shared__ float st[TILE * 3];       // 12 KB target tile
    __shared__ float seg_part[MAXG];
    __shared__ int   bnd[MAXG + 1];

    const int tid  = threadIdx.x;
    const int b    = blockIdx.x / tilesPerBatch;
    const int tile = blockIdx.x % tilesPerBatch;
    const int v0   = tile * TILE;
    const int count = min(TILE, N - v0);

    if (tid <= G) bnd[tid] = idx[b * (G + 1) + tid];
    if (tid < G)  seg_part[tid] = 0.0f;

    const char* gp = (const char*)(pred   + (size_t)b * N * 3 + (size_t)v0 * 3);
    const char* gt = (const char*)(target + (size_t)b * N * 3 + (size_t)v0 * 3);
    const int bytes = count * 12;

#if HAVE_ASYNC_LDS
    #pragma unroll
    for (int i = 0; i < (TILE * 12) / (16 * THREADS); ++i) {   // 3 b128 packets/thread/array
        int off = (tid + i * THREADS) * 16;
        if (off < bytes) {
            async_copy_b128(gp + off, (char*)sp + off);
            async_copy_b128(gt + off, (char*)st + off);
        }
    }
    asm volatile("s_wait_asynccnt 0" ::: "memory");
#else
    #pragma unroll
    for (int i = 0; i < (TILE * 12) / (16 * THREADS); ++i) {
        int off = (tid + i * THREADS) * 16;
        if (off < bytes) {
            *(float4*)((char*)sp + off) = *(const float4*)(gp + off);
            *(float4*)((char*)st + off) = *(const float4*)(gt + off);
        }
    }
#endif
    __syncthreads();

    // Each thread: 4 consecutive vertices; run-length flush into LDS atomics.
    int   g   = 0;
    int   cur = -1;
    float acc = 0.0f;
    const int base = tid * VPT;
    #pragma unroll
    for (int k = 0; k < VPT; ++k) {
        int v = base + k;
        if (v < count) {
            float dx = sp[v * 3 + 0] - st[v * 3 + 0];
            float dy = sp[v * 3 + 1] - st[v * 3 + 1];
            float dz = sp[v * 3 + 2] - st[v * 3 + 2];
            float e  = sqrtf(dx * dx + dy * dy + dz * dz);
            int vg = v0 + v;
            while (g + 1 < G && vg >= bnd[g + 1]) ++g;   // monotone boundary scan
            if (g != cur) {
                if (cur >= 0) atomicAdd(&seg_part[cur], acc);  // ds_add_f32
                cur = g; acc = 0.0f;
            }
            acc += e;
        }
    }
    if (cur >= 0) atomicAdd(&seg_part[cur], acc);
    __syncthreads();

    if (tid < G) {
        float s = seg_part[tid];
        if (s != 0.0f) atomicAdd(&seg_sum[b * G + tid], s);  // global_atomic_add_f32
    }
}

// ---------------- kernel 2: means, one-hot argmax, per-type average -------
__global__ void l2acc_finalize_kernel(const float* __restrict__ seg_sum,  // [B*G]
                                      const int*   __restrict__ idx,      // [B, G+1]
                                      const float* __restrict__ itype,    // [B*G, T]
                                      float*       __restrict__ out,      // [T]
                                      int B, int G, int T) {
    __shared__ float tsum[MAXT];
    __shared__ float tcnt[MAXT];
    const int tid = threadIdx.x;
    if (tid < T) { tsum[tid] = 0.0f; tcnt[tid] = 0.0f; }
    __syncthreads();

    const int BG = B * G;
    for (int i = tid; i < BG; i += blockDim.x) {
        int b = i / G, gg = i % G;
        int c0 = idx[b * (G + 1) + gg];
        int c1 = idx[b * (G + 1) + gg + 1];
        float mean = seg_sum[i] / (float)(c1 - c0);
        const float* row = itype + (size_t)i * T;
        int best = 0; float bv = row[0];
        for (int t = 1; t < T; ++t)               // first-max tie rule == jnp.argmax
            if (row[t] > bv) { bv = row[t]; best = t; }
        atomicAdd(&tsum[best], mean);
        atomicAdd(&tcnt[best], 1.0f);
    }
    __syncthreads();
    if (tid < T) out[tid] = (tcnt[tid] > 0.0f) ? (tsum[tid] / tcnt[tid]) : 0.0f;
}

// --------------------------------------------------------------------------
extern "C" void kernel_launch(void* const* d_in, const int* in_sizes, int n_in,
                              void* d_out, int out_size, void* d_ws, size_t ws_size,
                              hipStream_t stream) {
    const float* pred   = (const float*)d_in[0];
    const float* target = (const float*)d_in[1];
    const int*   idx    = (const int*)d_in[2];
    const float* itype  = (const float*)d_in[3];
    float* out     = (float*)d_out;
    float* seg_sum = (float*)d_ws;

    // Recover shapes (only D=3 hardcoded, per reference semantics):
    const int T  = out_size;                 // 5
    const int BG = in_sizes[3] / T;          // B*G = 256
    const int B  = in_sizes[2] - BG;         // B*(G+1) - B*G = 32
    const int G  = BG / B;                   // 8
    const int N  = in_sizes[0] / (3 * B);    // 200000
    const int tilesPerBatch = (N + TILE - 1) / TILE;

    l2acc_init_kernel<<<(BG + 255) / 256, 256, 0, stream>>>(seg_sum, BG);
    l2acc_seg_sum_kernel<<<B * tilesPerBatch, THREADS, 0, stream>>>(
        pred, target, idx, seg_sum, N, G, tilesPerBatch);
    l2acc_finalize_kernel<<<1, 256, 0, stream>>>(seg_sum, idx, itype, out, B, G, T);
}